// DilatedAvgPooling_5239860101152
// MI455X (gfx1250) — compile-verified
//
#include <hip/hip_runtime.h>

// Dilated (d=2) 3x3 zero-padded box mean over (16,256,128,128) fp32.
// Memory-bound (AI ~1.25 FLOP/B): optimize the data path, not FLOPs.
// One workgroup per (b,c) plane. TDM DMA (tensor_load_to_lds) pulls the
// 128x128 plane into a 132x132 LDS tile (pitch forced by TDM LDS-padding),
// halo ring is zeroed in parallel with the DMA, compute reads LDS via
// ds_load_b128, stores are 16B-coalesced global_store_b128.

#define WDIM   128
#define HDIM   128
#define PITCH  132                       // 128 + 2*2 halo
#define TILE_FLOATS (PITCH * PITCH)      // 17424 -> 69,696 bytes LDS

typedef unsigned int u32x4 __attribute__((ext_vector_type(4)));
typedef int          i32x8 __attribute__((ext_vector_type(8)));
typedef int          i32x4 __attribute__((ext_vector_type(4)));

__global__ void dil3x3_mean_kernel(const float* __restrict__ x,
                                   float* __restrict__ out)
{
    extern __shared__ __align__(16) float P[];   // 132x132 padded tile
    const int bc  = blockIdx.x;                  // fused (b, c): 0..4095
    const int tid = threadIdx.x;                 // 0..255 (8 waves of 32)

    const unsigned long long gaddr =
        (unsigned long long)(x + (size_t)bc * (HDIM * WDIM));

#if __has_builtin(__builtin_amdgcn_tensor_load_to_lds)
    // ---- wave 0 issues the plane DMA FIRST; halo zeroing overlaps it ----
    if (tid < 32) {
        // Low 32 bits of the flat shared address == byte offset in this
        // workgroup's LDS allocation (LDS aperture truncates to addr[31:0]).
        unsigned int lds_off =
            (unsigned int)(unsigned long long)(const void*)P;

        // ---- D# group 0 (128b): count=1 | lds_addr | global_addr | type=2
        u32x4 g0;
        g0[0] = 1u;                                        // count = 1 valid
        g0[1] = lds_off + (2 * PITCH + 2) * 4;             // interior (2,2)
        g0[2] = (unsigned int)(gaddr & 0xFFFFFFFFull);     // global_addr lo
        g0[3] = (unsigned int)((gaddr >> 32) & 0x01FFFFFFull)
              | (2u << 30);                                // type = 2 (image)

        // ---- D# group 1 (256b)
        // w0: mask=0 | data_size=2 (4B) | pad_enable |
        //     pad_interval=6 (128 DW) | pad_amount=3 (4 DW) => LDS pitch 132
        i32x8 g1;
        g1[0] = (int)((2u << 16) | (1u << 20) | (6u << 22) | (3u << 25));
        g1[1] = (int)(128u << 16);   // tensor_dim0[15:0]=128 @ bits[79:48] lo
        g1[2] = (int)(128u << 16);   // tensor_dim0 hi=0 | tensor_dim1 lo=128
        g1[3] = (int)(128u << 16);   // tensor_dim1 hi=0 | tile_dim0=128
        g1[4] = (int)(128u);         // tile_dim1=128 | tile_dim2=0 (2D)
        g1[5] = (int)(128u);         // tensor_dim0_stride = 128 (row pitch)
        g1[6] = (int)(16384u << 16); // dim0_stride hi=0 | dim1_stride lo=H*W
        g1[7] = 0;                   // dim1_stride hi (unused for 2D tile)

        i32x4 g2 = {0, 0, 0, 0};     // dims >2 unused
        i32x4 g3 = {0, 0, 0, 0};
        i32x8 g4 = {0, 0, 0, 0, 0, 0, 0, 0};  // unused group (VADDR4 = NULL)

        __builtin_amdgcn_tensor_load_to_lds(g0, g1, g2, g3, g4, /*cpol=*/0);
    }
#endif

    // ---- zero the 2-wide halo ring while the DMA is in flight ----
    // (DMA writes only the interior; halo addresses are disjoint.)
    // 132 x b128 stores (top/bottom rows) + 256 x b64 stores (side pairs).
    {
        const float4 z4 = make_float4(0.f, 0.f, 0.f, 0.f);
        const float2 z2 = make_float2(0.f, 0.f);
        for (int t = tid; t < 388; t += 256) {
            if (t < 132) {
                // rows 0..1 (t<66) and rows 130..131: 264 floats each region
                int base = (t < 66) ? (t * 4) : (130 * PITCH + (t - 66) * 4);
                *(float4*)&P[base] = z4;             // 16B-aligned (pitch 528B)
            } else {
                int u = t - 132;                     // 0..255
                int r = 2 + (u >> 1);                // rows 2..129
                int base = r * PITCH + ((u & 1) ? 130 : 0); // cols {0,1}/{130,131}
                *(float2*)&P[base] = z2;             // 8B-aligned
            }
        }
    }

#if __has_builtin(__builtin_amdgcn_tensor_load_to_lds)
    if (tid < 32) {
#if __has_builtin(__builtin_amdgcn_s_wait_tensorcnt)
        __builtin_amdgcn_s_wait_tensorcnt(0);
#else
        asm volatile("s_wait_tensorcnt 0" ::: "memory");
#endif
    }
#else
    // Fallback: cooperative vector loads into the LDS interior.
    for (int t = tid; t < (HDIM * WDIM) / 4; t += 256) {
        int r = t >> 5;              // 0..127
        int c = (t & 31) * 4;        // 0..124 step 4
        const float4 v =
            *(const float4*)(x + (size_t)bc * (HDIM * WDIM) + r * WDIM + c);
        float* d = &P[(r + 2) * PITCH + (c + 2)];
        d[0] = v.x; d[1] = v.y; d[2] = v.z; d[3] = v.w;
    }
#endif
    __syncthreads();   // halo zeros (DScnt) + DMA tile now visible to all waves

    // ---- compute: each thread -> 4 consecutive cols x 16 row-iterations ----
    const int c0 = (tid & 31) * 4;   // output col base (16B aligned in LDS)
    const int r0 = tid >> 5;         // 0..7
    float* o = out + (size_t)bc * (HDIM * WDIM);
    const float k = 1.0f / 9.0f;

    #pragma unroll
    for (int it = 0; it < 16; ++it) {
        const int r = it * 8 + r0;   // output row 0..127
        // taps at padded rows r, r+2, r+4 ; padded cols c0 .. c0+7
        const float* p0 = &P[(r    ) * PITCH + c0];
        const float* p1 = &P[(r + 2) * PITCH + c0];
        const float* p2 = &P[(r + 4) * PITCH + c0];
        float4 a0 = *(const float4*)p0, b0 = *(const float4*)(p0 + 4);
        float4 a1 = *(const float4*)p1, b1 = *(const float4*)(p1 + 4);
        float4 a2 = *(const float4*)p2, b2 = *(const float4*)(p2 + 4);

        float s0 = (a0.x + a0.z + b0.x) + (a1.x + a1.z + b1.x) + (a2.x + a2.z + b2.x);
        float s1 = (a0.y + a0.w + b0.y) + (a1.y + a1.w + b1.y) + (a2.y + a2.w + b2.y);
        float s2 = (a0.z + b0.x + b0.z) + (a1.z + b1.x + b1.z) + (a2.z + b2.x + b2.z);
        float s3 = (a0.w + b0.y + b0.w) + (a1.w + b1.y + b1.w) + (a2.w + b2.y + b2.w);

        *(float4*)(o + (size_t)r * WDIM + c0) =
            make_float4(s0 * k, s1 * k, s2 * k, s3 * k);
    }
}

extern "C" void kernel_launch(void* const* d_in, const int* in_sizes, int n_in,
                              void* d_out, int out_size, void* d_ws, size_t ws_size,
                              hipStream_t stream) {
    (void)n_in; (void)out_size; (void)d_ws; (void)ws_size;
    const float* x  = (const float*)d_in[0];
    float* out      = (float*)d_out;

    const int planes = in_sizes[0] / (HDIM * WDIM);        // B*C = 4096
    const size_t shmem = (size_t)TILE_FLOATS * sizeof(float); // 69,696 B

    // Opt-in to >64KB dynamic LDS (no-op if already allowed; not a stream op).
    (void)hipFuncSetAttribute((const void*)dil3x3_mean_kernel,
                              hipFuncAttributeMaxDynamicSharedMemorySize,
                              (int)shmem);

    dil3x3_mean_kernel<<<planes, 256, shmem, stream>>>(x, out);
}